// GRID_18863496364434
// MI455X (gfx1250) — compile-verified
//
#include <hip/hip_runtime.h>
#include <cstdint>

#define C_   32
#define D_   128
#define H_   128
#define W_   128
#define DHW_ (D_ * H_ * W_)

// Broadcast lane `src` of a wave-uniform-index shuffle via v_readlane (SGPR,
// no LDS/bpermute traffic). `src` must be wave-uniform (loop counter here).
__device__ __forceinline__ float bcast(float v, int src)
{
    return __uint_as_float(__builtin_amdgcn_readlane(__float_as_uint(v), src));
}

// -----------------------------------------------------------------------------
// Kernel 1: layout transform  vol[C][D][H][W]  ->  volt[D][H][W][C]
// Makes the 32 channel values of one voxel a single 128B cacheline, so each
// trilinear corner in the sampler is one fully-consumed coalesced wave32 load.
// Tile staged in LDS with CDNA5 async global->LDS copies (ASYNCcnt path).
// -----------------------------------------------------------------------------
__global__ __launch_bounds__(256) void vol_transpose(const float* __restrict__ vol,
                                                     float* __restrict__ volt)
{
    __shared__ float tile[32][33];               // +1 pad: conflict-free transpose
    const int tx  = threadIdx.x & 31;            // 0..31
    const int tyq = threadIdx.x >> 5;            // 0..7
    const int tileId   = blockIdx.x;             // D_*H_*(W_/32) tiles
    const int w0       = (tileId & 3) * 32;      // W_/32 == 4
    const int dh       = tileId >> 2;            // d*H_ + h
    const int spatBase = dh * W_ + w0;

    // 32(ch) x 32(w) tile straight into LDS: async global->LDS, no VGPR trip.
#pragma unroll
    for (int i = 0; i < 4; ++i) {
        const int c = tyq + 8 * i;
        unsigned ldsOff = (unsigned)(uintptr_t)&tile[c][tx];        // LDS byte offset
        int      gOff   = (c * DHW_ + spatBase + tx) * 4;           // byte offset < 2^31
        asm volatile("global_load_async_to_lds_b32 %0, %1, %2"
                     :
                     : "v"(ldsOff), "v"(gOff), "s"(vol)
                     : "memory");
    }
    asm volatile("s_wait_asynccnt 0" ::: "memory");
    __syncthreads();

    // Write transposed: consecutive lanes (tx = channel) -> contiguous bytes.
#pragma unroll
    for (int i = 0; i < 4; ++i) {
        const int w = tyq + 8 * i;
        volt[(size_t)(spatBase + w) * C_ + tx] = tile[tx][w];
    }
}

// One point, channel-innermost volume: lane == channel.
// All index math is wave-uniform -> forced scalar (SGPR); the 8 corner loads
// are scalar-base + lane-offset coalesced 128B wave loads.
__device__ __forceinline__ float sample_point_t(const float* __restrict__ volt,
                                                int lane, float cz, float cy, float cx)
{
    const float uz = (cz + 1.0f) * 63.5f;        // (c+1)*0.5*(128-1)
    const float uy = (cy + 1.0f) * 63.5f;
    const float ux = (cx + 1.0f) * 63.5f;
    const float fz = floorf(uz), fy = floorf(uy), fx = floorf(ux);
    const float tz = uz - fz, ty = uy - fy, txf = ux - fx;

    const int iz = __builtin_amdgcn_readfirstlane((int)fz);
    const int iy = __builtin_amdgcn_readfirstlane((int)fy);
    const int ix = __builtin_amdgcn_readfirstlane((int)fx);

    // Axis weights with border validity folded in (padding_mode='zeros').
    const float wz0 = (iz     >= 0 && iz     < D_) ? (1.f - tz)  : 0.f;
    const float wz1 = (iz + 1 >= 0 && iz + 1 < D_) ? tz          : 0.f;
    const float wy0 = (iy     >= 0 && iy     < H_) ? (1.f - ty)  : 0.f;
    const float wy1 = (iy + 1 >= 0 && iy + 1 < H_) ? ty          : 0.f;
    const float wx0 = (ix     >= 0 && ix     < W_) ? (1.f - txf) : 0.f;
    const float wx1 = (ix + 1 >= 0 && ix + 1 < W_) ? txf         : 0.f;

    // Clamped scalar indices (matches reference's clip-then-mask).
    const int iz0 = iz     < 0 ? 0 : (iz     > D_ - 1 ? D_ - 1 : iz);
    const int iz1 = iz + 1 < 0 ? 0 : (iz + 1 > D_ - 1 ? D_ - 1 : iz + 1);
    const int iy0 = iy     < 0 ? 0 : (iy     > H_ - 1 ? H_ - 1 : iy);
    const int iy1 = iy + 1 < 0 ? 0 : (iy + 1 > H_ - 1 ? H_ - 1 : iy + 1);
    const int ix0 = ix     < 0 ? 0 : (ix     > W_ - 1 ? W_ - 1 : ix);
    const int ix1 = ix + 1 < 0 ? 0 : (ix + 1 > W_ - 1 ? W_ - 1 : ix + 1);

    const int rz0 = iz0 * (H_ * W_ * C_), rz1 = iz1 * (H_ * W_ * C_);
    const int ry0 = iy0 * (W_ * C_),      ry1 = iy1 * (W_ * C_);
    const int rx0 = ix0 * C_,             rx1 = ix1 * C_;

    const float v000 = volt[rz0 + ry0 + rx0 + lane];
    const float v001 = volt[rz0 + ry0 + rx1 + lane];
    const float v010 = volt[rz0 + ry1 + rx0 + lane];
    const float v011 = volt[rz0 + ry1 + rx1 + lane];
    const float v100 = volt[rz1 + ry0 + rx0 + lane];
    const float v101 = volt[rz1 + ry0 + rx1 + lane];
    const float v110 = volt[rz1 + ry1 + rx0 + lane];
    const float v111 = volt[rz1 + ry1 + rx1 + lane];

    const float w00 = wz0 * wy0, w01 = wz0 * wy1;
    const float w10 = wz1 * wy0, w11 = wz1 * wy1;
    float acc;
    acc  = v000 * (w00 * wx0);
    acc += v001 * (w00 * wx1);
    acc += v010 * (w01 * wx0);
    acc += v011 * (w01 * wx1);
    acc += v100 * (w10 * wx0);
    acc += v101 * (w10 * wx1);
    acc += v110 * (w11 * wx0);
    acc += v111 * (w11 * wx1);
    return acc;
}

// -----------------------------------------------------------------------------
// Kernel 2: sampler. One wave32 per 32 points; each lane pre-loads one point's
// coords (non-temporal: read-once), broadcast via v_readlane. Output stores are
// non-temporal so the 256MB write-once stream doesn't evict the volume from
// the 192MB L2 (the gather hit-rate is the whole game at 23.3 TB/s HBM).
// -----------------------------------------------------------------------------
__global__ __launch_bounds__(256) void sample_t(const float* __restrict__ coords,
                                                const float* __restrict__ volt,
                                                float* __restrict__ out, int N)
{
    const int lane = threadIdx.x & 31;
    const int wave = (blockIdx.x * 256 + (int)threadIdx.x) >> 5;
    const int pb   = wave << 5;                  // 32 points per wave
    if (pb >= N) return;

    const int myp = pb + lane;
    float c0 = 0.f, c1 = 0.f, c2 = 0.f;
    if (myp < N) {
        c0 = __builtin_nontemporal_load(&coords[3 * myp + 0]);   // -> D (z)
        c1 = __builtin_nontemporal_load(&coords[3 * myp + 1]);   // -> H (y)
        c2 = __builtin_nontemporal_load(&coords[3 * myp + 2]);   // -> W (x)
    }

    if (N - pb >= 32) {
        // Fast path: compile-time trip count, unrolled for memory-level parallelism.
#pragma unroll 4
        for (int p = 0; p < 32; ++p) {
            const float acc = sample_point_t(volt, lane,
                                             bcast(c0, p), bcast(c1, p), bcast(c2, p));
            __builtin_nontemporal_store(acc, &out[(size_t)(pb + p) * C_ + lane]);
        }
    } else {
        const int npts = N - pb;
        for (int p = 0; p < npts; ++p) {
            const float acc = sample_point_t(volt, lane,
                                             bcast(c0, p), bcast(c1, p), bcast(c2, p));
            __builtin_nontemporal_store(acc, &out[(size_t)(pb + p) * C_ + lane]);
        }
    }
}

// -----------------------------------------------------------------------------
// Fallback (ws too small for the 256MB transposed volume): sample the original
// [C,D,H,W] layout directly; lane == channel with per-lane channel base.
// -----------------------------------------------------------------------------
__global__ __launch_bounds__(256) void sample_direct(const float* __restrict__ coords,
                                                     const float* __restrict__ vol,
                                                     float* __restrict__ out, int N)
{
    const int lane = threadIdx.x & 31;
    const int wave = (blockIdx.x * 256 + (int)threadIdx.x) >> 5;
    const int pb   = wave << 5;
    if (pb >= N) return;

    const int myp = pb + lane;
    float c0 = 0.f, c1 = 0.f, c2 = 0.f;
    if (myp < N) {
        c0 = __builtin_nontemporal_load(&coords[3 * myp + 0]);
        c1 = __builtin_nontemporal_load(&coords[3 * myp + 1]);
        c2 = __builtin_nontemporal_load(&coords[3 * myp + 2]);
    }
    const int npts = (N - pb < 32) ? (N - pb) : 32;
    const size_t chanBase = (size_t)lane * DHW_;

    for (int p = 0; p < npts; ++p) {
        const float uz = (bcast(c0, p) + 1.0f) * 63.5f;
        const float uy = (bcast(c1, p) + 1.0f) * 63.5f;
        const float ux = (bcast(c2, p) + 1.0f) * 63.5f;
        const float fz = floorf(uz), fy = floorf(uy), fx = floorf(ux);
        const float tz = uz - fz, ty = uy - fy, txf = ux - fx;
        const int iz = __builtin_amdgcn_readfirstlane((int)fz);
        const int iy = __builtin_amdgcn_readfirstlane((int)fy);
        const int ix = __builtin_amdgcn_readfirstlane((int)fx);

        const float wz0 = (iz     >= 0 && iz     < D_) ? (1.f - tz)  : 0.f;
        const float wz1 = (iz + 1 >= 0 && iz + 1 < D_) ? tz          : 0.f;
        const float wy0 = (iy     >= 0 && iy     < H_) ? (1.f - ty)  : 0.f;
        const float wy1 = (iy + 1 >= 0 && iy + 1 < H_) ? ty          : 0.f;
        const float wx0 = (ix     >= 0 && ix     < W_) ? (1.f - txf) : 0.f;
        const float wx1 = (ix + 1 >= 0 && ix + 1 < W_) ? txf         : 0.f;

        const int iz0 = iz     < 0 ? 0 : (iz     > D_ - 1 ? D_ - 1 : iz);
        const int iz1 = iz + 1 < 0 ? 0 : (iz + 1 > D_ - 1 ? D_ - 1 : iz + 1);
        const int iy0 = iy     < 0 ? 0 : (iy     > H_ - 1 ? H_ - 1 : iy);
        const int iy1 = iy + 1 < 0 ? 0 : (iy + 1 > H_ - 1 ? H_ - 1 : iy + 1);
        const int ix0 = ix     < 0 ? 0 : (ix     > W_ - 1 ? W_ - 1 : ix);
        const int ix1 = ix + 1 < 0 ? 0 : (ix + 1 > W_ - 1 ? W_ - 1 : ix + 1);

        const int rz0 = iz0 * (H_ * W_), rz1 = iz1 * (H_ * W_);
        const int ry0 = iy0 * W_,        ry1 = iy1 * W_;

        const float v000 = vol[chanBase + (rz0 + ry0 + ix0)];
        const float v001 = vol[chanBase + (rz0 + ry0 + ix1)];
        const float v010 = vol[chanBase + (rz0 + ry1 + ix0)];
        const float v011 = vol[chanBase + (rz0 + ry1 + ix1)];
        const float v100 = vol[chanBase + (rz1 + ry0 + ix0)];
        const float v101 = vol[chanBase + (rz1 + ry0 + ix1)];
        const float v110 = vol[chanBase + (rz1 + ry1 + ix0)];
        const float v111 = vol[chanBase + (rz1 + ry1 + ix1)];

        const float w00 = wz0 * wy0, w01 = wz0 * wy1;
        const float w10 = wz1 * wy0, w11 = wz1 * wy1;
        float acc;
        acc  = v000 * (w00 * wx0);
        acc += v001 * (w00 * wx1);
        acc += v010 * (w01 * wx0);
        acc += v011 * (w01 * wx1);
        acc += v100 * (w10 * wx0);
        acc += v101 * (w10 * wx1);
        acc += v110 * (w11 * wx0);
        acc += v111 * (w11 * wx1);

        __builtin_nontemporal_store(acc, &out[(size_t)(pb + p) * C_ + lane]);
    }
}

extern "C" void kernel_launch(void* const* d_in, const int* in_sizes, int n_in,
                              void* d_out, int out_size, void* d_ws, size_t ws_size,
                              hipStream_t stream)
{
    const float* coords = (const float*)d_in[0];   // [N,3] f32
    const float* vol    = (const float*)d_in[1];   // [C,D,H,W] f32
    float*       out    = (float*)d_out;           // [N,C] f32

    const int N = in_sizes[0] / 3;
    const int nWaves  = (N + 31) / 32;
    const int threads = nWaves * 32;
    const int blocks  = (threads + 255) / 256;

    const size_t need = (size_t)DHW_ * C_ * sizeof(float);   // 256 MB
    if (ws_size >= need) {
        float* volt = (float*)d_ws;
        vol_transpose<<<D_ * H_ * (W_ / 32), 256, 0, stream>>>(vol, volt);
        sample_t<<<blocks, 256, 0, stream>>>(coords, volt, out, N);
    } else {
        sample_direct<<<blocks, 256, 0, stream>>>(coords, vol, out, N);
    }
}